// BiquadCoeffFilter_31628139167986
// MI455X (gfx1250) — compile-verified
//
#include <hip/hip_runtime.h>
#include <math.h>

// ---------------- problem constants ----------------
#define B      32          // batch rows
#define NS     262144      // samples per row
#define NCP    256         // coefficient control points
#define KC     5           // logits per control point
#define LCH    128         // chunk length (samples per lane)
#define TC     128         // chunks per block == blockDim.x
#define TILE   (LCH*TC)    // 16384 samples staged per block (64KB)
#define TPR    (NS/TILE)   // tiles per row = 16
#define CPR    (NS/LCH)    // chunks per row = 2048
#define PADL   (LCH+1)     // padded LDS stride in dwords (TDM inserts 1 dword per 128)
#define EPSF   0.001f
#define STAB   (1.0f-EPSF)
#define POSCALE ((float)(NCP-1)/(float)(NS-1))

typedef unsigned int u32x4 __attribute__((ext_vector_type(4)));
typedef int          i32x8 __attribute__((ext_vector_type(8)));
typedef int          i32x4 __attribute__((ext_vector_type(4)));

// ---------------- TDM: async DMA of a contiguous 16384-dword tile into LDS,
// with hardware pad insertion of 1 dword every 128 dwords (bank-conflict-free
// strided reads: lane j reads lds[j*129 + t]). ----------------
__device__ __forceinline__ void stage_tile(const float* __restrict__ gsrc,
                                           float* lds_dst, int tid) {
#if __has_builtin(__builtin_amdgcn_tensor_load_to_lds) && __has_builtin(__builtin_amdgcn_s_wait_tensorcnt)
  if (tid == 0) {   // TDM ignores EXEC; issue from one wave only
    unsigned long long ga = (unsigned long long)(uintptr_t)gsrc;
    // flat LDS aperture: low 32 bits == LDS byte offset
    unsigned lds_off = (unsigned)(uintptr_t)lds_dst;
    u32x4 g0;
    g0[0] = 1u;                                              // count=1 valid descriptor
    g0[1] = lds_off;                                         // lds_addr
    g0[2] = (unsigned)ga;                                    // global_addr[31:0]
    g0[3] = (unsigned)((ga >> 32) & 0x01FFFFFFull) | (2u << 30); // addr[56:32] | type=2
    i32x8 g1;
    g1[0] = (int)((2u << 16) | (1u << 20) | (6u << 22));     // data_size=4B, pad_enable, interval=128 dwords, pad=1 dword
    g1[1] = (int)(((unsigned)TILE & 0xFFFFu) << 16);         // tensor_dim0[15:0] in bits 63:48
    g1[2] = (int)((((unsigned)TILE >> 16) & 0xFFFFu) | (1u << 16)); // tensor_dim0 hi | tensor_dim1=1
    g1[3] = (int)(((unsigned)TILE & 0xFFFFu) << 16);         // tile_dim0 = 16384 dwords
    g1[4] = 1;                                               // tile_dim1 = 1
    g1[5] = TILE;                                            // tensor_dim0_stride (lo)
    g1[6] = 0;
    g1[7] = 0;
    i32x4 z4 = {0, 0, 0, 0};
    i32x8 z8 = {0, 0, 0, 0, 0, 0, 0, 0};
    __builtin_amdgcn_tensor_load_to_lds(g0, g1, z4, z4, z8, 0);
    __builtin_amdgcn_s_wait_tensorcnt(0);
  }
#else
  // fallback: coalesced manual copy with identical padded layout
  for (int g = tid; g < TILE; g += TC)
    lds_dst[g + (g >> 7)] = gsrc[g];
#endif
}

// ---------------- cached control-point segment ----------------
// A 128-sample chunk spans only ~0.125 in interpolation position, so the
// segment index changes at most once per chunk: keep base+delta in registers
// and reload from LDS only on transition.
struct CoeffSeg {
  float c0[KC];
  float d[KC];
  int   i0;
  __device__ __forceinline__ void load(const float* lds_cl, int i) {
    i0 = i;
    int i1 = min(i + 1, NCP - 1);
    const float* p0 = lds_cl + i  * KC;
    const float* p1 = lds_cl + i1 * KC;
#pragma unroll
    for (int k = 0; k < KC; ++k) {
      c0[k] = p0[k];
      d[k]  = p1[k] - p0[k];
    }
  }
};

__device__ __forceinline__ void coeffs_from_seg(CoeffSeg& seg, const float* lds_cl,
                                                int t_abs,
                                                float& a1, float& a2,
                                                float& b0, float& b1, float& b2) {
  float pos = (float)t_abs * POSCALE;
  int   i0  = (int)pos;
  if (i0 != seg.i0) seg.load(lds_cl, i0);   // at most once per 128-sample chunk
  float w = pos - (float)i0;
  float l0 = seg.c0[0] + seg.d[0] * w;
  float l1 = seg.c0[1] + seg.d[1] * w;
  b0 = seg.c0[2] + seg.d[2] * w;
  b1 = seg.c0[3] + seg.d[3] * w;
  b2 = seg.c0[4] + seg.d[4] * w;
  a1 = 2.0f * tanhf(l0) * STAB;
  float aa = fabsf(a1);
  a2 = 0.5f * ((2.0f - aa) * tanhf(l1) * STAB + aa);
}

// ---------------- pass 1: per-chunk affine summaries (zero-IC scan) ----------------
__global__ __launch_bounds__(TC) void lpc_pass1(const float* __restrict__ x,
                                                const float* __restrict__ cl,
                                                float* __restrict__ summ) {
  __shared__ float lds_x[TC * PADL];   // 16512 dwords (64KB + pads)
  __shared__ float lds_cl[NCP * KC];   // 1280 dwords
  const int row  = blockIdx.x / TPR;
  const int tile = blockIdx.x % TPR;
  const int tid  = threadIdx.x;

  const float* clr = cl + (size_t)row * NCP * KC;
  for (int i = tid; i < NCP * KC; i += TC) lds_cl[i] = clr[i];

  stage_tile(x + (size_t)row * NS + (size_t)tile * TILE, lds_x, tid);
  __syncthreads();

  float u1 = 1.0f, u2 = 0.0f;   // response to y[-1]=1
  float v1 = 0.0f, v2 = 1.0f;   // response to y[-2]=1
  float p1 = 0.0f, p2 = 0.0f;   // particular (zero ICs)
  const float* lx    = lds_x + tid * PADL;
  const int    tbase = (tile * TC + tid) * LCH;

  CoeffSeg seg;
  seg.load(lds_cl, (int)((float)tbase * POSCALE));

#pragma unroll 2
  for (int t = 0; t < LCH; ++t) {
    float a1, a2, b0, b1, b2;
    coeffs_from_seg(seg, lds_cl, tbase + t, a1, a2, b0, b1, b2);
    float xt = lx[t];
    float u = -a1 * u1 - a2 * u2;
    float v = -a1 * v1 - a2 * v2;
    float p = xt - a1 * p1 - a2 * p2;
    u2 = u1; u1 = u;
    v2 = v1; v1 = v;
    p2 = p1; p1 = p;
  }
  const int ci = row * CPR + tile * TC + tid;
  float* s = summ + (size_t)ci * 6;
  s[0] = u1; s[1] = v1; s[2] = p1;
  s[3] = u2; s[4] = v2; s[5] = p2;
}

// ---------------- pass 2 of scan: serial chunk-boundary combine ----------------
__global__ __launch_bounds__(256) void lpc_combine(const float* __restrict__ summ,
                                                   float* __restrict__ states) {
  const int W = 64;                      // chunks per window
  __shared__ float lds_s[B * W * 6];     // 48KB
  const int tid = threadIdx.x;
  float s1 = 0.0f, s2 = 0.0f;
  for (int w = 0; w < CPR / W; ++w) {
    for (int idx = tid; idx < B * W * 6; idx += 256) {
      int r   = idx / (W * 6);
      int rem = idx % (W * 6);
      lds_s[idx] = summ[((size_t)r * CPR + (size_t)w * W) * 6 + rem];
    }
    __syncthreads();
    if (tid < B) {
      const int row = tid;
      for (int c = 0; c < W; ++c) {
        const float* e = lds_s + row * (W * 6) + c * 6;
        const int ci = row * CPR + w * W + c;
        states[(size_t)ci * 2 + 0] = s1;          // exact incoming (y[t0-1], y[t0-2])
        states[(size_t)ci * 2 + 1] = s2;
        float ns1 = e[0] * s1 + e[1] * s2 + e[2];
        float ns2 = e[3] * s1 + e[4] * s2 + e[5];
        s1 = ns1; s2 = ns2;
      }
    }
    __syncthreads();
  }
}

// ---------------- pass 3: exact re-scan + fused time-varying FIR ----------------
__global__ __launch_bounds__(TC) void lpc_pass2(const float* __restrict__ x,
                                                const float* __restrict__ cl,
                                                const float* __restrict__ states,
                                                float* __restrict__ out) {
  __shared__ float lds_x[TC * PADL];
  __shared__ float lds_cl[NCP * KC];
  const int row  = blockIdx.x / TPR;
  const int tile = blockIdx.x % TPR;
  const int tid  = threadIdx.x;

  const float* clr = cl + (size_t)row * NCP * KC;
  for (int i = tid; i < NCP * KC; i += TC) lds_cl[i] = clr[i];

  stage_tile(x + (size_t)row * NS + (size_t)tile * TILE, lds_x, tid);
  __syncthreads();

  const int ci = row * CPR + tile * TC + tid;
  float s1 = states[(size_t)ci * 2 + 0];   // y[t-1]
  float s2 = states[(size_t)ci * 2 + 1];   // y[t-2]
  float* ly = lds_x + tid * PADL;          // in-place: x[t] -> out[t]
  const int tbase = (tile * TC + tid) * LCH;

  CoeffSeg seg;
  seg.load(lds_cl, (int)((float)tbase * POSCALE));

#pragma unroll 2
  for (int t = 0; t < LCH; ++t) {
    float a1, a2, b0, b1, b2;
    coeffs_from_seg(seg, lds_cl, tbase + t, a1, a2, b0, b1, b2);
    float xt = ly[t];
    float y  = xt - a1 * s1 - a2 * s2;
    ly[t]    = b0 * y + b1 * s1 + b2 * s2;  // FIR taps are previous IIR outputs
    s2 = s1; s1 = y;
  }
  __syncthreads();

  float* od = out + (size_t)row * NS + (size_t)tile * TILE;
  for (int g = tid; g < TILE; g += TC)     // coalesced, conflict-free un-padding
    od[g] = lds_x[g + (g >> 7)];
}

// ---------------- launcher ----------------
extern "C" void kernel_launch(void* const* d_in, const int* in_sizes, int n_in,
                              void* d_out, int out_size, void* d_ws, size_t ws_size,
                              hipStream_t stream) {
  const float* x  = (const float*)d_in[0];   // (32, 262144) f32
  const float* cl = (const float*)d_in[1];   // (32, 256, 5) f32
  float* out = (float*)d_out;

  float* summ   = (float*)d_ws;                         // B*CPR*6 floats (1.5 MB)
  float* states = summ + (size_t)B * CPR * 6;           // B*CPR*2 floats (0.5 MB)

  lpc_pass1  <<<B * TPR, TC,  0, stream>>>(x, cl, summ);
  lpc_combine<<<1,        256, 0, stream>>>(summ, states);
  lpc_pass2  <<<B * TPR, TC,  0, stream>>>(x, cl, states, out);
}